// ScatterDiagonal1_40656160424525
// MI455X (gfx1250) — compile-verified
//
#include <hip/hip_runtime.h>

// CDNA5 (gfx1250) wave32 WMMA kernel for per-diagonal linear layers.
// Y_i = X_i @ W_i^T + b_i  scattered into out(B,9,9,128).

typedef __attribute__((ext_vector_type(2))) float v2f;
typedef __attribute__((ext_vector_type(8))) float v8f;
typedef __attribute__((ext_vector_type(2))) int   v2i;

#define CIN  128
#define COUT 128

// Async global->LDS staging (gfx1250 GLOBAL_LOAD_ASYNC_TO_LDS_B64) if the
// toolchain declares the builtin; otherwise fall back to a synchronous fill.
#if defined(__has_builtin)
#if __has_builtin(__builtin_amdgcn_global_load_async_to_lds_b64)
#define USE_ASYNC_LDS 1
#endif
#endif

// Builtin signature (from the round-3 diagnostic): param0 is a pointer to a
// 2 x i32 vector in AS(1) (printed as "__device__" in HIP mode); param1 is the
// LDS-side counterpart in AS(3).
typedef __attribute__((address_space(1))) v2i* gv2i_ptr;
typedef __attribute__((address_space(3))) v2i* sv2i_ptr;

// 256 threads = 8 waves. Block handles 256 rows of the flattened (B*L, CIN)
// input; each wave computes a 32(M) x 128(N) strip as TWO 16-row M-tiles that
// share each B fragment in registers (2 WMMA per DS fragment load).
// Accumulate over K=128 in 32 steps of V_WMMA_F32_16X16X4_F32.
__global__ __launch_bounds__(256)
void diag_linear_wmma(const float* __restrict__ x,
                      const float* __restrict__ w,
                      const float* __restrict__ bias,
                      float* __restrict__ out,
                      int L, int pos_base)
{
    // W staged in LDS, pre-swizzled into B-fragment order:
    // frag[((t*32 + kq)*32 + lane)] = { W[16t+nn][4kq+2h], W[16t+nn][4kq+2h+1] }
    // where nn = lane&15, h = lane>>4.  Exactly 64 KB, lane-fastest so the
    // inner-loop ds_load_b64 reads are bank-conflict free.
    __shared__ float Wl[CIN * COUT];

    const int tid  = threadIdx.x;
    const int wave = tid >> 5;       // 0..7
    const int lane = tid & 31;
    const int half = lane >> 4;      // 0/1
    const int nn   = lane & 15;      // 0..15

    v2f* Wf = (v2f*)Wl;

    // ---- cooperative swizzled fill of W (4096 v2f chunks / 256 threads) ----
    #pragma unroll
    for (int j = 0; j < 16; ++j) {
        int idx = tid + j * 256;          // destination fragment index
        int lp  = idx & 31;
        int kq  = (idx >> 5) & 31;
        int t   = idx >> 10;
        int row = t * 16 + (lp & 15);     // Cout row of W
        int col = kq * 4 + (lp >> 4) * 2; // Cin column (8B aligned)
#ifdef USE_ASYNC_LDS
        // 8-byte async DMA chunk: global (AS1) -> LDS (AS3), no VGPR data path.
        __builtin_amdgcn_global_load_async_to_lds_b64(
            (gv2i_ptr)(w + row * CIN + col), (sv2i_ptr)(Wf + idx), 0, 0);
#else
        Wf[idx] = *(const v2f*)(w + row * CIN + col);
#endif
    }
#ifdef USE_ASYNC_LDS
    asm volatile("s_wait_asynccnt 0" ::: "memory");
#endif
    __syncthreads();

    const unsigned mBase = blockIdx.x * 256u + (unsigned)wave * 32u;

    // ---- accumulators initialised with bias (bias depends only on N) ----
    v8f acc0[8], acc1[8];
    #pragma unroll
    for (int t = 0; t < 8; ++t) {
        float bv = bias[t * 16 + nn];
        v8f bvv = (v8f){bv, bv, bv, bv, bv, bv, bv, bv};
        acc0[t] = bvv;
        acc1[t] = bvv;
    }

    // A-fragment sources: lane nn walks its own X row; 8-byte loads.
    const float* xrow0 = x + (size_t)(mBase + (unsigned)nn) * CIN + 2 * half;
    const float* xrow1 = xrow0 + 16 * CIN;

    // ---- K loop: 32 steps x (8 N-tiles x 2 M-tiles) = 512 WMMAs per wave ----
    #pragma unroll 2
    for (int kq = 0; kq < 32; ++kq) {
        v2f a0 = *(const v2f*)(xrow0 + 4 * kq);
        v2f a1 = *(const v2f*)(xrow1 + 4 * kq);
        #pragma unroll
        for (int t = 0; t < 8; ++t) {
            v2f b = Wf[t * 1024 + kq * 32 + lane];
            acc0[t] = __builtin_amdgcn_wmma_f32_16x16x4_f32(
                false, a0, false, b, (short)0, acc0[t], false, false);
            acc1[t] = __builtin_amdgcn_wmma_f32_16x16x4_f32(
                false, a1, false, b, (short)0, acc1[t], false, false);
        }
    }

    // ---- scatter stores ----
    // C/D layout: VGPR j, lane -> element (M = j + 8*half, N = 16*t + nn).
    // Flattened row r = b*L + l maps to out offset (b*81 + pos_base + 8*l)*128,
    // using pos = r*9 + c = 8*(rmin + l) + i  (closed form, no index table).
    #pragma unroll
    for (int m = 0; m < 2; ++m) {
        const v8f* acc = m ? acc1 : acc0;
        #pragma unroll
        for (int j = 0; j < 8; ++j) {
            unsigned row_g = mBase + 16u * (unsigned)m + (unsigned)(half * 8 + j);
            unsigned bb = row_g / (unsigned)L;
            unsigned l  = row_g - bb * (unsigned)L;
            float* rowp = out + ((size_t)bb * 81u + (unsigned)pos_base + 8u * l) * COUT;
            #pragma unroll
            for (int t = 0; t < 8; ++t) {
                __builtin_nontemporal_store(acc[t][j], rowp + t * 16 + nn);
            }
        }
    }
}

extern "C" void kernel_launch(void* const* d_in, const int* in_sizes, int n_in,
                              void* d_out, int out_size, void* d_ws, size_t ws_size,
                              hipStream_t stream)
{
    const float* weights = (const float*)d_in[0];  // (17, 128, 128)
    const float* bias    = (const float*)d_in[1];  // (17, 128)
    float* out = (float*)d_out;                    // (8192, 9, 9, 128), fully overwritten

    static const int LENS[17] = {1, 2, 3, 4, 5, 6, 7, 8, 9, 8, 7, 6, 5, 4, 3, 2, 1};

    for (int i = 0; i < 17; ++i) {
        int L = LENS[i];
        int rmin = (i > 8) ? (i - 8) : 0;
        int pos_base = 8 * rmin + i;               // pos(l) = pos_base + 8*l
        const float* x = (const float*)d_in[2 + i];
        unsigned grid = 32u * (unsigned)L;         // (8192*L)/256 blocks
        diag_linear_wmma<<<dim3(grid), dim3(256), 0, stream>>>(
            x,
            weights + (size_t)i * CIN * COUT,
            bias + (size_t)i * COUT,
            out, L, pos_base);
    }
}